// Painting_7576322310696
// MI455X (gfx1250) — compile-verified
//
#include <hip/hip_runtime.h>
#include <stdint.h>

// Alpha-compositing scan: canvas = canvas + a*(poly - canvas), a = poly[3]*0.8
// Memory-bound streamer (512 MB read once, ~22 us floor at 23.3 TB/s).
// CDNA5 path: deep prefetch via global_load_async_to_lds_b128 (ASYNCcnt) into
// a 4-stage LDS ring, consumed with ds_load_b128. Each thread reads back only
// the LDS it wrote -> no barriers, only split wait counters.

#define Hh 512
#define Ww 512
#define HW (Hh * Ww)
#define LAYER_STRIDE (4 * HW)                    // elements per layer
#define LAYER_BYTES ((uint64_t)LAYER_STRIDE * 4) // 4 MiB
#define CHAN_PLANE_BYTES ((uint64_t)HW * 4)      // 1 MiB
#define OPACITY 0.8f
#define BLOCK 256
#define PIX_PER_THREAD 4                         // one float4 per channel
#define PIX_PER_BLOCK (BLOCK * PIX_PER_THREAD)   // 1024 pixels
#define STAGES 4                                 // LDS ring depth
#define CHAN_BYTES (BLOCK * 16)                  // 4096 B per channel slab
#define STAGE_BYTES (4 * CHAN_BYTES)             // 16384 B per stage

#define WAIT_ASYNC(imm) asm volatile("s_wait_asynccnt " #imm ::: "memory")
#define WAIT_DS0()      asm volatile("s_wait_dscnt 0" ::: "memory")

__global__ __launch_bounds__(BLOCK)
void composite_scan_kernel(const float* __restrict__ polys,
                           float* __restrict__ out, int nlayers) {
    // [stage][chan][tid] float4 tiles: 4*4*256*16 = 64 KB LDS
    __shared__ float4 smem[STAGES * 4 * BLOCK];

    const int tid = threadIdx.x;
    const int p0  = blockIdx.x * PIX_PER_BLOCK + tid * PIX_PER_THREAD;
    const uint32_t voff = (uint32_t)(p0 * 4);    // byte offset in channel plane

    // Low 32 bits of a flat pointer into __shared__ == LDS byte offset
    // (LDS aperture is 4GB-aligned; ADDR[31:0] is the LDS address).
    const uint32_t lds_base = (uint32_t)(uintptr_t)(&smem[tid]);

    // One stage = 4 async b128 loads (one per channel plane) from gbase.
    auto issue_stage = [&](uint64_t gbase, int stage) {
#pragma unroll
        for (int c = 0; c < 4; ++c) {
            uint32_t lds_addr = lds_base + (uint32_t)(stage * STAGE_BYTES + c * CHAN_BYTES);
            uint64_t sbase    = gbase + (uint64_t)c * CHAN_PLANE_BYTES;
            // GVS mode: mem = SGPR64 base + VGPR32 offset; dest = per-lane LDS addr
            asm volatile("global_load_async_to_lds_b128 %0, %1, %2"
                         :: "v"(lds_addr), "v"(voff), "s"(sbase)
                         : "memory");
        }
    };

    float4 acc0 = make_float4(1.f, 1.f, 1.f, 1.f);  // R plane (4 pixels)
    float4 acc1 = acc0, acc2 = acc0, acc3 = acc0;   // G, B, A planes

    // Consume stage `st`: pull 4 channel tiles from LDS, composite into acc.
    auto consume = [&](int st) {
        const float4* sp = &smem[st * 4 * BLOCK + tid];
        float4 v0 = sp[0 * BLOCK];
        float4 v1 = sp[1 * BLOCK];
        float4 v2 = sp[2 * BLOCK];
        float4 v3 = sp[3 * BLOCK];                  // alpha channel
        float4 a;
        a.x = v3.x * OPACITY; a.y = v3.y * OPACITY;
        a.z = v3.z * OPACITY; a.w = v3.w * OPACITY;
#define COMPOSITE(ac, vv)                      \
        ac.x = fmaf(a.x, vv.x - ac.x, ac.x);   \
        ac.y = fmaf(a.y, vv.y - ac.y, ac.y);   \
        ac.z = fmaf(a.z, vv.z - ac.z, ac.z);   \
        ac.w = fmaf(a.w, vv.w - ac.w, ac.w);
        COMPOSITE(acc0, v0)
        COMPOSITE(acc1, v1)
        COMPOSITE(acc2, v2)
        COMPOSITE(acc3, v3)
#undef COMPOSITE
    };

    // ---- Prime: fill all 4 stages (16 async loads outstanding). ----
    uint64_t gb = (uint64_t)(uintptr_t)polys;       // advances one layer per issue
#pragma unroll
    for (int s = 0; s < STAGES; ++s) {
        issue_stage(gb, s);
        if (s + 1 < nlayers) gb += LAYER_BYTES;     // clamp for tiny N
    }

    // ---- Main loop: steady 16 outstanding; <=12 frees the oldest stage. ----
    const int main_iters = (nlayers > STAGES) ? (nlayers - STAGES) : 0;
    int n = 0;
#pragma unroll 4
    for (; n < main_iters; ++n) {
        WAIT_ASYNC(12);
        const int st = n & (STAGES - 1);
        consume(st);
        WAIT_DS0();                 // our ds_loads retired before overwrite
        issue_stage(gb, st);        // layer n + STAGES
        gb += LAYER_BYTES;
    }

    // ---- Drain: 4 remaining stages, descending immediate waits. ----
    if (n < nlayers) { WAIT_ASYNC(12); consume(n & 3); ++n; }
    if (n < nlayers) { WAIT_ASYNC(8);  consume(n & 3); ++n; }
    if (n < nlayers) { WAIT_ASYNC(4);  consume(n & 3); ++n; }
    if (n < nlayers) { WAIT_ASYNC(0);  consume(n & 3); ++n; }

    // Store canvas: out layout (1,4,H,W); p0 is 16B-aligned.
    *(float4*)(out + 0 * HW + p0) = acc0;
    *(float4*)(out + 1 * HW + p0) = acc1;
    *(float4*)(out + 2 * HW + p0) = acc2;
    *(float4*)(out + 3 * HW + p0) = acc3;
}

extern "C" void kernel_launch(void* const* d_in, const int* in_sizes, int n_in,
                              void* d_out, int out_size, void* d_ws, size_t ws_size,
                              hipStream_t stream) {
    const float* polys = (const float*)d_in[0];
    int nlayers = in_sizes[0] / (4 * Hh * Ww);      // 128 for reference shapes
    dim3 grid(HW / PIX_PER_BLOCK);                  // 256 blocks x 256 threads
    composite_scan_kernel<<<grid, BLOCK, 0, stream>>>(polys, (float*)d_out, nlayers);
}